// GATLayer_Hetersum_19971597926647
// MI455X (gfx1250) — compile-verified
//
#include <hip/hip_runtime.h>
#include <hip/hip_bf16.h>
#include <math.h>

typedef float v2f __attribute__((ext_vector_type(2)));
typedef float v8f __attribute__((ext_vector_type(8)));

#define IN_F  128
#define OUT_F 64
#define NEG_SLOPE 0.01f

// ---- monotone ordered-int encoding of float (for atomicMax on signed int) ----
__device__ __forceinline__ int f2ord(float f) {
    int i = __float_as_int(f);
    return (i >= 0) ? i : (i ^ 0x7FFFFFFF);
}
__device__ __forceinline__ float ord2f(int i) {
    return __int_as_float((i >= 0) ? i : (i ^ 0x7FFFFFFF));
}

// ---------------------------------------------------------------------------
// init: zero out[N*64], denom[N], m_int[N] = ordered(-inf)
// ---------------------------------------------------------------------------
__global__ void k_init(float* __restrict__ out, float* __restrict__ denom,
                       int* __restrict__ m_int, int N) {
    int i = blockIdx.x * blockDim.x + threadIdx.x;
    if (i < N * OUT_F) out[i] = 0.0f;
    if (i < N) {
        denom[i] = 0.0f;
        m_int[i] = f2ord(-INFINITY);
    }
}

// ---------------------------------------------------------------------------
// GEMM: z[N,64] = h[N,128] @ W[128,64], fp32 WMMA 16x16x4.
// One wave computes a 16-row x 64-col tile (4 N-tiles of 16).
// W staged in LDS as K-pairs: sWp[kp*64 + c] = {W[2kp][c], W[2kp+1][c]}
// so every B fragment is one aligned ds_load_b64 (no regrouping movs).
// ---------------------------------------------------------------------------
__global__ __launch_bounds__(256) void k_gemm(const float* __restrict__ h,
                                              const float* __restrict__ W,
                                              float* __restrict__ z, int N) {
    __shared__ __align__(16) v2f sWp[(IN_F / 2) * OUT_F];  // 64*64 float2 = 32 KB

    const int tid = threadIdx.x;
    // cooperative fill: element i = (kp, c); global reads coalesced over c
    for (int i = tid; i < (IN_F / 2) * OUT_F; i += 256) {
        const int kp = i >> 6;
        const int c  = i & 63;
        v2f v;
        v.x = W[(2 * kp + 0) * OUT_F + c];
        v.y = W[(2 * kp + 1) * OUT_F + c];
        sWp[i] = v;
    }
    __syncthreads();

    const int wave = tid >> 5;     // 0..7
    const int lane = tid & 31;
    const int half = lane >> 4;    // 0 or 1
    const int m    = lane & 15;

    const int row0 = (blockIdx.x * 8 + wave) * 16;
    if (row0 >= N) return;  // wave-uniform: EXEC stays all-ones for active waves

    int rowA = row0 + m;
    if (rowA >= N) rowA = N - 1;   // clamp so WMMA lanes stay valid
    const float* __restrict__ hrow = h + (size_t)rowA * IN_F;

    v8f acc0 = {}, acc1 = {}, acc2 = {}, acc3 = {};

#pragma unroll 4
    for (int kk = 0; kk < IN_F / 4; ++kk) {
        // this lane-half covers K = 4*kk + 2*half, +1  ->  K-pair index:
        const int kp = kk * 2 + half;
        // A fragment: h[row][4kk+2half], h[row][4kk+2half+1] -> one b64 load
        v2f A = *(const v2f*)(hrow + kk * 4 + 2 * half);
        // B fragments: one b64 LDS load per N-tile, already paired
        const v2f* __restrict__ Brow = sWp + kp * OUT_F + m;
        v2f B0 = Brow[0];
        v2f B1 = Brow[16];
        v2f B2 = Brow[32];
        v2f B3 = Brow[48];

        acc0 = __builtin_amdgcn_wmma_f32_16x16x4_f32(false, A, false, B0, (short)0, acc0, false, false);
        acc1 = __builtin_amdgcn_wmma_f32_16x16x4_f32(false, A, false, B1, (short)0, acc1, false, false);
        acc2 = __builtin_amdgcn_wmma_f32_16x16x4_f32(false, A, false, B2, (short)0, acc2, false, false);
        acc3 = __builtin_amdgcn_wmma_f32_16x16x4_f32(false, A, false, B3, (short)0, acc3, false, false);
    }

    // D layout: VGPR r, lanes 0-15 -> M=r, lanes 16-31 -> M=r+8; N = tile*16 + m
#pragma unroll
    for (int r = 0; r < 8; ++r) {
        const int row = row0 + r + 8 * half;
        if (row < N) {
            float* zr = z + (size_t)row * OUT_F + m;
            zr[ 0] = acc0[r];
            zr[16] = acc1[r];
            zr[32] = acc2[r];
            zr[48] = acc3[r];
        }
    }
}

// ---------------------------------------------------------------------------
// scores: s_src[n] = z[n]·a[0:64], s_dst[n] = z[n]·a[64:128]
// one wave32 per node, float2 per lane, shuffle reduction
// ---------------------------------------------------------------------------
__global__ void k_scores(const float* __restrict__ z, const float* __restrict__ a,
                         float* __restrict__ s_src, float* __restrict__ s_dst, int N) {
    const int gwave = (blockIdx.x * blockDim.x + threadIdx.x) >> 5;
    const int lane  = threadIdx.x & 31;
    if (gwave >= N) return;

    float2 zv = ((const float2*)(z + (size_t)gwave * OUT_F))[lane];
    float2 a1 = ((const float2*)a)[lane];
    float2 a2 = ((const float2*)(a + OUT_F))[lane];
    float p1 = zv.x * a1.x + zv.y * a1.y;
    float p2 = zv.x * a2.x + zv.y * a2.y;
#pragma unroll
    for (int off = 16; off > 0; off >>= 1) {
        p1 += __shfl_xor(p1, off, 32);
        p2 += __shfl_xor(p2, off, 32);
    }
    if (lane == 0) {
        s_src[gwave] = p1;
        s_dst[gwave] = p2;
    }
}

// ---------------------------------------------------------------------------
// edge pass 1: e = leaky_relu(s_src[src]+s_dst[dst]); segment max via
// ordered-int atomicMax
// ---------------------------------------------------------------------------
__global__ void k_edge_max(const float* __restrict__ s_src, const float* __restrict__ s_dst,
                           const int* __restrict__ src, const int* __restrict__ dst,
                           float* __restrict__ e, int* __restrict__ m_int, int E) {
    int i = blockIdx.x * blockDim.x + threadIdx.x;
    if (i >= E) return;
    float v = s_src[src[i]] + s_dst[dst[i]];
    v = (v > 0.0f) ? v : NEG_SLOPE * v;
    e[i] = v;
    atomicMax(&m_int[dst[i]], f2ord(v));
}

// ---------------------------------------------------------------------------
// edge pass 2: ex = exp(e - m[dst]) (in place), denom[dst] += ex
// ---------------------------------------------------------------------------
__global__ void k_edge_exp(const int* __restrict__ dst, const int* __restrict__ m_int,
                           float* __restrict__ e, float* __restrict__ denom, int E) {
    int i = blockIdx.x * blockDim.x + threadIdx.x;
    if (i >= E) return;
    int d = dst[i];
    float ex = __expf(e[i] - ord2f(m_int[d]));
    e[i] = ex;
    atomicAdd(&denom[d], ex);
}

// ---------------------------------------------------------------------------
// scatter: out[dst] += (ex/denom[dst]) * z[src]; 32 lanes per edge,
// float2 gather per lane, 2 coalesced atomic f32 adds per lane
// ---------------------------------------------------------------------------
__global__ void k_scatter(const float* __restrict__ z, const int* __restrict__ src,
                          const int* __restrict__ dst, const float* __restrict__ ex,
                          const float* __restrict__ denom, float* __restrict__ out, int E) {
    const int eidx = (blockIdx.x * blockDim.x + threadIdx.x) >> 5;
    const int lane = threadIdx.x & 31;
    if (eidx >= E) return;
    const int s = src[eidx];
    const int d = dst[eidx];
    const float alpha = ex[eidx] / denom[d];
    float2 zv = ((const float2*)(z + (size_t)s * OUT_F))[lane];
    float* o = out + (size_t)d * OUT_F + 2 * lane;
    atomicAdd(o + 0, alpha * zv.x);
    atomicAdd(o + 1, alpha * zv.y);
}

// ---------------------------------------------------------------------------
extern "C" void kernel_launch(void* const* d_in, const int* in_sizes, int n_in,
                              void* d_out, int out_size, void* d_ws, size_t ws_size,
                              hipStream_t stream) {
    const float* h = (const float*)d_in[0];
    const float* W = (const float*)d_in[1];
    const float* a = (const float*)d_in[2];
    const int* src = (const int*)d_in[3];
    const int* dst = (const int*)d_in[4];

    const int N = in_sizes[0] / IN_F;
    const int E = in_sizes[3];

    float* out = (float*)d_out;

    // workspace carve-up (all 4-byte types)
    float* z     = (float*)d_ws;                 // N*64
    float* s_src = z + (size_t)N * OUT_F;        // N
    float* s_dst = s_src + N;                    // N
    float* denom = s_dst + N;                    // N
    int*   m_int = (int*)(denom + N);            // N
    float* e     = (float*)(m_int + N);          // E

    // 1) init out / denom / m
    {
        int total = N * OUT_F;
        k_init<<<(total + 255) / 256, 256, 0, stream>>>(out, denom, m_int, N);
    }
    // 2) z = h @ W  (fp32 WMMA)
    {
        int rowTiles = (N + 15) / 16;
        k_gemm<<<(rowTiles + 7) / 8, 256, 0, stream>>>(h, W, z, N);
    }
    // 3) attention scores
    k_scores<<<(N * 32 + 255) / 256, 256, 0, stream>>>(z, a, s_src, s_dst, N);
    // 4) edge score + segment max
    k_edge_max<<<(E + 255) / 256, 256, 0, stream>>>(s_src, s_dst, src, dst, e, m_int, E);
    // 5) exp + denom
    k_edge_exp<<<(E + 255) / 256, 256, 0, stream>>>(dst, m_int, e, denom, E);
    // 6) weighted scatter-sum
    {
        long long threads = (long long)E * 32;
        int blocks = (int)((threads + 255) / 256);
        k_scatter<<<blocks, 256, 0, stream>>>(z, src, dst, e, denom, out, E);
    }
}